// MVGeometricProduct_22462678958651
// MI455X (gfx1250) — compile-verified
//
#include <hip/hip_runtime.h>
#include <hip/hip_bf16.h>
#include <math.h>

// ---------------- Clifford algebra tables (D=3, NB=8) ----------------
// blade order sorted by (grade, value): [0,1,2,4,3,5,6,7]
constexpr int BLADE[8] = {0, 1, 2, 4, 3, 5, 6, 7};
constexpr int GRADE[8] = {0, 1, 1, 1, 2, 2, 2, 3};
// k index for pair t = i*8+j : index of blade[i]^blade[j]
constexpr int PAIR_K[64] = {
    0,1,2,3,4,5,6,7,
    1,0,4,5,2,3,7,6,
    2,4,0,6,1,7,3,5,
    3,5,6,0,7,1,2,4,
    4,2,1,7,0,6,5,3,
    5,3,7,1,6,0,4,2,
    6,7,3,2,5,4,0,1,
    7,6,5,4,3,2,1,0};
// lexicographic index of valid grade path (gi,gj,gk) -> [0,20); idx = gi*16+gj*4+gk
constexpr int PATHLUT[64] = {
     0, 0, 0, 0,  0, 1, 0, 0,  0, 0, 2, 0,  0, 0, 0, 3,
     0, 4, 0, 0,  5, 0, 6, 0,  0, 7, 0, 8,  0, 0, 9, 0,
     0, 0,10, 0,  0,11, 0,12, 13, 0,14, 0,  0,15, 0, 0,
     0, 0, 0,16,  0, 0,17, 0,  0,18, 0, 0, 19, 0, 0, 0};

#define F_ 768
#define B_ 1024
#define NPATH 20
#define FT 64    // features per block
#define BT 4     // batches per loop iteration
#define ITER 4   // batch iterations per block (amortizes LDS staging)

// ---------------- Precompute: per-feature signed weight table -------------
// W64Q layout: [fblk][t>>2][f&63][t&3]  (4096 floats = 16KB per fblk chunk)
// SIGT layout: [fblk][g][f&63]          (256 floats per fblk chunk)
__global__ __launch_bounds__(256) void build_tables(
    const float* __restrict__ weight, const float* __restrict__ a,
    float* __restrict__ W64Q, float* __restrict__ SIGT) {
    int gid = blockIdx.x * 256 + threadIdx.x;   // 0 .. F_*64-1
    int f = gid >> 6;
    int t = gid & 63;
    int i = t >> 3, j = t & 7;
    int bi = BLADE[i], bj = BLADE[j];
    int k = PAIR_K[t];
    // reorder sign: for 3-bit blades cnt = popc((bi>>1)&bj) + popc((bi>>2)&bj)
    int cnt = __popc((bi >> 1) & bj) + __popc((bi >> 2) & bj);
    float s = (cnt & 1) ? -1.0f : 1.0f;
    int p = PATHLUT[GRADE[i] * 16 + GRADE[j] * 4 + GRADE[k]];
    float val = s * weight[f * NPATH + p];
    int fblk = f >> 6, fl = f & 63;
    W64Q[fblk * 4096 + (t >> 2) * 256 + fl * 4 + (t & 3)] = val;
    if (t < 4) {
        SIGT[fblk * 256 + t * 64 + fl] = 1.0f / (1.0f + expf(-a[f * 4 + t]));
    }
}

// ---------------- Main kernel: memory-bound geometric product -------------
__global__ __launch_bounds__(256) void gp_main(
    const float* __restrict__ v, const float* __restrict__ w,
    const float* __restrict__ W64Q, const float* __restrict__ SIGT,
    float* __restrict__ out) {
    __shared__ __align__(16) float sW[FT * 64];   // 16 KB weight tile
    __shared__ __align__(16) float sSig[FT * 4];  // 1 KB sigmoid tile

    const int NFB = F_ / FT;  // 12
    int fblk = blockIdx.x % NFB;
    int bgrp = blockIdx.x / NFB;   // 0 .. B_/(BT*ITER)-1
    int tid = threadIdx.x;

    // --- CDNA5 async copy: stage weight tile global -> LDS (ASYNCcnt path)
    {
        const float* gsrc = W64Q + (size_t)fblk * (FT * 64);
        unsigned ldsW = (unsigned)(size_t)(&sW[0]);
#pragma unroll
        for (int pass = 0; pass < 4; ++pass) {
            unsigned off = (unsigned)(pass * 256 + tid) * 16u;  // b128 per lane
            unsigned lds = ldsW + off;
            asm volatile("global_load_async_to_lds_b128 %0, %1, %2"
                         :: "v"(lds), "v"(off), "s"(gsrc)
                         : "memory");
        }
        if (tid < 64) {  // waves 0,1 fully active
            const float* gs2 = SIGT + (size_t)fblk * (FT * 4);
            unsigned off = (unsigned)tid * 16u;
            unsigned lds = (unsigned)(size_t)(&sSig[0]) + off;
            asm volatile("global_load_async_to_lds_b128 %0, %1, %2"
                         :: "v"(lds), "v"(off), "s"(gs2)
                         : "memory");
        }
        asm volatile("s_wait_asynccnt 0" ::: "memory");
    }
    __syncthreads();

    int fl = tid & 63;       // consecutive lanes -> consecutive features (coalesced)
    int bl = tid >> 6;       // 0..3
    int f = fblk * FT + fl;

    // Hoist this feature's 64 signed weights LDS -> VGPRs (bank-conflict-free b128)
    float wt[64];
    const float4* sW4 = (const float4*)sW;
#pragma unroll
    for (int tq = 0; tq < 16; ++tq) {
        float4 q = sW4[tq * 64 + fl];
        wt[tq * 4 + 0] = q.x;
        wt[tq * 4 + 1] = q.y;
        wt[tq * 4 + 2] = q.z;
        wt[tq * 4 + 3] = q.w;
    }
    float s0 = sSig[0 * 64 + fl];
    float s1 = sSig[1 * 64 + fl];
    float s2 = sSig[2 * 64 + fl];
    float s3 = sSig[3 * 64 + fl];

    for (int iter = 0; iter < ITER; ++iter) {
        int b = bgrp * (BT * ITER) + iter * BT + bl;
        size_t base = ((size_t)b * F_ + f) * 8;

        float4 va = *(const float4*)(v + base);
        float4 vb = *(const float4*)(v + base + 4);
        float4 wa = *(const float4*)(w + base);
        float4 wb = *(const float4*)(w + base + 4);

        // per-grade norms: g0={0}, g1={1,2,3}, g2={4,5,6}, g3={7}
        float n0 = fabsf(wa.x);
        float n1 = sqrtf(wa.y * wa.y + wa.z * wa.z + wa.w * wa.w);
        float n2 = sqrtf(wb.x * wb.x + wb.y * wb.y + wb.z * wb.z);
        float n3 = fabsf(wb.w);
        float d0 = 1.0f / (s0 * (n0 - 1.0f) + 1.0f);
        float d1 = 1.0f / (s1 * (n1 - 1.0f) + 1.0f);
        float d2 = 1.0f / (s2 * (n2 - 1.0f) + 1.0f);
        float d3 = 1.0f / (s3 * (n3 - 1.0f) + 1.0f);

        float wn[8] = {wa.x * d0, wa.y * d1, wa.z * d1, wa.w * d1,
                       wb.x * d2, wb.y * d2, wb.z * d2, wb.w * d3};
        float vv[8] = {va.x, va.y, va.z, va.w, vb.x, vb.y, vb.z, vb.w};
        float acc[8] = {0.f, 0.f, 0.f, 0.f, 0.f, 0.f, 0.f, 0.f};

        // fully unrolled 64-term Cayley contraction; compile-time i/j/k indices.
#pragma unroll
        for (int t = 0; t < 64; ++t) {
            acc[t & 7] = fmaf(vv[t >> 3] * wt[t], wn[PAIR_K[t]], acc[t & 7]);
        }

        float4 oa = {acc[0], acc[1], acc[2], acc[3]};
        float4 ob = {acc[4], acc[5], acc[6], acc[7]};
        *(float4*)(out + base) = oa;
        *(float4*)(out + base + 4) = ob;
    }
}

// ---------------- Launch ----------------
extern "C" void kernel_launch(void* const* d_in, const int* in_sizes, int n_in,
                              void* d_out, int out_size, void* d_ws, size_t ws_size,
                              hipStream_t stream) {
    const float* v      = (const float*)d_in[0];   // (B,F,8)
    const float* w      = (const float*)d_in[1];   // (B,F,8)
    const float* weight = (const float*)d_in[2];   // (F,20)
    const float* a      = (const float*)d_in[3];   // (F,4)
    float* out = (float*)d_out;                    // (B,F,8)

    float* W64Q = (float*)d_ws;          // F*64 floats
    float* SIGT = W64Q + F_ * 64;        // F*4 floats

    build_tables<<<(F_ * 64) / 256, 256, 0, stream>>>(weight, a, W64Q, SIGT);

    const int nblocks = (F_ / FT) * (B_ / (BT * ITER));  // 12 * 64 = 768
    gp_main<<<nblocks, 256, 0, stream>>>(v, w, W64Q, SIGT, out);
}